// VDPFirstConv_50783693308284
// MI455X (gfx1250) — compile-verified
//
#include <hip/hip_runtime.h>
#include <stdint.h>

typedef __attribute__((ext_vector_type(2))) float v2f;
typedef __attribute__((ext_vector_type(4))) float v4f;
typedef __attribute__((ext_vector_type(8))) float v8f;

#define KSZ   7
#define KK    49            // 7*7 fan-in
#define KPAD  52            // K padded to multiple of 4 (rows stay 8B aligned)
#define FOUT  64
#define HIN   224
#define WIN   224
#define HOUT  218
#define WOUT  218
#define NPIX  (HOUT*WOUT)   // 47524
#define NBATCH 64
#define MU_SZ ((size_t)NBATCH*FOUT*NPIX)
#define NTILES ((NPIX + 15)/16)   // 2971 pixel tiles of 16
#define WAVES_PER_BLOCK 8
#define THREADS (WAVES_PER_BLOCK*32)

// Patch offset for kernel element k (clamped into the valid 7x7 range; the
// clamped elements are value-masked to zero so the clamp address is harmless).
__device__ __host__ constexpr int off_of(int k) {
    int kc = (k <= KK - 1) ? k : (KK - 1);
    return (kc / KSZ) * WIN + (kc % KSZ);
}

// Fused conv (WMMA f32 GEMM, M=filters, N=pixels, K=49->52) + variance path.
__global__ __launch_bounds__(THREADS)
void vdp_conv_wmma_kernel(const float* __restrict__ mu_in,
                          const float* __restrict__ w_mu,
                          const float* __restrict__ w_sigma,
                          float* __restrict__ out) {
    __shared__ float sW[FOUT*KPAD];   // zero-padded weight rows, 13.0 KB
    __shared__ float sSig2[FOUT];     // softplus(w_sigma)

    const int tid = threadIdx.x;

    // Stage the 64x49 weights straight into the padded LDS layout with
    // per-lane async memory->LDS gathers (ASYNCcnt path, no VGPR round-trip).
    for (int g = tid; g < FOUT*KK; g += THREADS) {
        const int f = g / KK, k = g - f*KK;
        const uint32_t ldsoff = (uint32_t)(uintptr_t)&sW[f*KPAD + k];
        const float*   gaddr  = w_mu + g;
        asm volatile("global_load_async_to_lds_b32 %0, %1, off"
                     :: "v"(ldsoff), "v"(gaddr) : "memory");
    }
    // Zero the 64*3 K-pad slots with plain DS stores.
    if (tid < FOUT*(KPAD-KK)) {
        const int f = tid / (KPAD-KK), k = KK + (tid - f*(KPAD-KK));
        sW[f*KPAD + k] = 0.0f;
    }
    if (tid < FOUT) sSig2[tid] = log1pf(expf(w_sigma[tid]));
    asm volatile("s_wait_asynccnt 0" ::: "memory");
    __syncthreads();

    const int wave = tid >> 5;
    const int lane = tid & 31;
    const int tile = blockIdx.x * WAVES_PER_BLOCK + wave;
    if (tile >= NTILES) return;
    const int b = blockIdx.y;

    const int  col   = lane & 15;         // pixel col (B/D) or filter row (A)
    const bool hi    = (lane >> 4) != 0;  // lanes 16-31 hold K = k0+2,k0+3
    const int  khalf = hi ? 2 : 0;

    // Pixel handled by this lane (clamped for the ragged last tile; stores masked).
    const int  p_lane = tile*16 + col;
    const bool pvalid = p_lane < NPIX;
    const int  p  = pvalid ? p_lane : (NPIX - 1);
    const int  ho = p / WOUT;
    const int  wo = p - ho*WOUT;
    const float* inb = mu_in + (size_t)b*HIN*WIN + ho*WIN + wo;

    // Pull the 7 input rows of this wave's patch region toward the caches.
    #pragma unroll
    for (int r = 0; r < KSZ; ++r)
        __builtin_prefetch(inb + r*WIN, 0, 0);

    // Per-lane base into the LDS weight tile: all A-fragment loads below use
    // compile-time instruction offsets off this single pointer.
    const float* wbase = &sW[col*KPAD + khalf];

    v8f acc0 = {}, acc1 = {}, acc2 = {}, acc3 = {};
    float ss = 0.0f;   // per-lane partial sum of squares of the patch

    #pragma unroll
    for (int ks = 0; ks < KPAD/4; ++ks) {
        // Compile-time patch offsets; lane-half select is a single cndmask.
        const int offa = hi ? off_of(4*ks + 2) : off_of(4*ks + 0);
        const int offb = hi ? off_of(4*ks + 3) : off_of(4*ks + 1);
        float va = inb[offa];
        float vb = inb[offb];
        if (ks == KPAD/4 - 1) {       // k = 48..51: only lo-half k=48 is real
            va = hi ? 0.0f : va;
            vb = 0.0f;
        }
        ss += va*va + vb*vb;

        v2f bfrag; bfrag.x = va; bfrag.y = vb;          // B: 4x16 patch block

        // A: 16x4 weight blocks, ds_load_b64 with immediate offsets.
        v2f a0 = *(const v2f*)(wbase + 0*16*KPAD + 4*ks);
        v2f a1 = *(const v2f*)(wbase + 1*16*KPAD + 4*ks);
        v2f a2 = *(const v2f*)(wbase + 2*16*KPAD + 4*ks);
        v2f a3 = *(const v2f*)(wbase + 3*16*KPAD + 4*ks);

        acc0 = __builtin_amdgcn_wmma_f32_16x16x4_f32(false, a0, false, bfrag, (short)0, acc0, false, false);
        acc1 = __builtin_amdgcn_wmma_f32_16x16x4_f32(false, a1, false, bfrag, (short)0, acc1, false, false);
        acc2 = __builtin_amdgcn_wmma_f32_16x16x4_f32(false, a2, false, bfrag, (short)0, acc2, false, false);
        acc3 = __builtin_amdgcn_wmma_f32_16x16x4_f32(false, a3, false, bfrag, (short)0, acc3, false, false);
    }

    // Combine the two K-halves of each pixel's sum of squares; /49 = x_dim.
    const float patch_ss = (ss + __shfl_xor(ss, 16)) * (1.0f/49.0f);

    if (!pvalid) return;

    const int rowSel = hi ? 8 : 0;     // D: lanes>=16 hold rows M=j+8
    float* __restrict__ mu_out  = out;
    float* __restrict__ sig_out = out + MU_SZ;

    v8f accs[4] = {acc0, acc1, acc2, acc3};
    #pragma unroll
    for (int mt = 0; mt < 4; ++mt) {
        const int fbase = mt*16 + rowSel;
        v4f slo = *(const v4f*)&sSig2[fbase];
        v4f shi = *(const v4f*)&sSig2[fbase + 4];
        #pragma unroll
        for (int j = 0; j < 8; ++j) {
            const int f = fbase + j;
            const size_t idx = ((size_t)(b*FOUT + f))*NPIX + p_lane;
            mu_out[idx]  = accs[mt][j];
            const float sg = (j < 4) ? slo[j] : shi[j-4];
            sig_out[idx] = patch_ss * sg;
        }
    }
}

// KL(N(w_mu, softplus(w_sigma) I) || N(0, I)), summed over the 64 kernels.
__global__ __launch_bounds__(64)
void vdp_kl_kernel(const float* __restrict__ w_mu,
                   const float* __restrict__ w_sigma,
                   float* __restrict__ out_kl) {
    __shared__ float red[FOUT];
    const int f = threadIdx.x;
    float s2 = log1pf(expf(w_sigma[f]));
    float musq = 0.0f;
    #pragma unroll
    for (int k = 0; k < KK; ++k) {
        float w = w_mu[f*KK + k];
        musq += w*w;
    }
    red[f] = (float)KK*s2 + musq - (float)KK - (float)KK*logf(s2);
    __syncthreads();
    if (f == 0) {
        float t = 0.0f;
        for (int i = 0; i < FOUT; ++i) t += red[i];
        *out_kl = 0.5f * t;
    }
}

extern "C" void kernel_launch(void* const* d_in, const int* in_sizes, int n_in,
                              void* d_out, int out_size, void* d_ws, size_t ws_size,
                              hipStream_t stream) {
    const float* mu_in   = (const float*)d_in[0];
    const float* w_mu    = (const float*)d_in[1];
    const float* w_sigma = (const float*)d_in[2];
    float* out = (float*)d_out;

    dim3 grid((NTILES + WAVES_PER_BLOCK - 1) / WAVES_PER_BLOCK, NBATCH);
    vdp_conv_wmma_kernel<<<grid, THREADS, 0, stream>>>(mu_in, w_mu, w_sigma, out);
    vdp_kl_kernel<<<1, 64, 0, stream>>>(w_mu, w_sigma, out + 2*MU_SZ);
}